// AQ_MLP_84799834293038
// MI455X (gfx1250) — compile-verified
//
#include <hip/hip_runtime.h>
#include <hip/hip_bf16.h>

typedef float v2f __attribute__((ext_vector_type(2)));
typedef float v8f __attribute__((ext_vector_type(8)));

#define BSZ 4096
#define NEC 512   // codebook size

// ---------------------------------------------------------------------------
// WMMA fp32 GEMM:  C[M,N] = act(A[M,K] @ W[N,K]^T)
// A, W row-major with K contiguous (lda = ldb = K), ldc = N.
// Each wave computes NT adjacent 16x16 tiles (16 x 16*NT strip): one A
// fragment feeds NT independent accumulators -> no single WMMA RAW chain and
// A traffic amortized NT x. 8 waves / 256-thread block -> 128 rows per block.
// f32 16x16x4 fragments: lane<16 holds K=k0..k0+1, lane>=16 holds K=k0+2..k0+3
// -> each operand fragment is a single b64 load per lane.
// ---------------------------------------------------------------------------
template<bool RELU, int NT, int NVALID>
__global__ __launch_bounds__(256)
void gemm_wmma_f32(const float* __restrict__ A, const float* __restrict__ W,
                   float* __restrict__ C, int M, int N, int K) {
  const int wave = threadIdx.x >> 5;
  const int lane = threadIdx.x & 31;
  const int r    = lane & 15;
  const int kh   = lane >> 4;                 // which K-pair this half-wave owns
  const int m0   = (blockIdx.y * 8 + wave) * 16;
  const int n0   = blockIdx.x * (16 * NT);

  const float* arow = A + (size_t)(m0 + r) * K + 2 * kh;
  const float* brow[NT];
#pragma unroll
  for (int t = 0; t < NT; ++t) {
    int nr = n0 + t * 16 + ((NVALID < 16 && r >= NVALID) ? 0 : r);
    brow[t] = W + (size_t)nr * K + 2 * kh;
  }

  v8f acc[NT];
#pragma unroll
  for (int t = 0; t < NT; ++t)
    acc[t] = (v8f){0.f, 0.f, 0.f, 0.f, 0.f, 0.f, 0.f, 0.f};

#pragma unroll 2
  for (int k0 = 0; k0 < K; k0 += 4) {
    v2f a = *(const v2f*)(arow + k0);
    v2f b[NT];
#pragma unroll
    for (int t = 0; t < NT; ++t) {
      b[t] = *(const v2f*)(brow[t] + k0);
      if (NVALID < 16 && r >= NVALID) { b[t].x = 0.f; b[t].y = 0.f; }
    }
#pragma unroll
    for (int t = 0; t < NT; ++t)
      acc[t] = __builtin_amdgcn_wmma_f32_16x16x4_f32(false, a, false, b[t],
                                                     (short)0, acc[t],
                                                     false, false);
  }

  if (r < NVALID) {
#pragma unroll
    for (int t = 0; t < NT; ++t) {
      const int col = n0 + t * 16 + r;
#pragma unroll
      for (int v = 0; v < 8; ++v) {
        const int row = m0 + v + 8 * kh;      // C layout: VGPR v, lane-half -> M
        float val = acc[t][v];
        if (RELU) val = fmaxf(val, 0.f);
        C[(size_t)row * N + col] = val;
      }
    }
  }
  (void)M;
}

// ---------------------------------------------------------------------------
// block-wide sum + scaled atomic accumulate (blockDim.x == 256)
// ---------------------------------------------------------------------------
__device__ inline void block_reduce_add(float val, float* target, float scale) {
  __shared__ float red[256];
  red[threadIdx.x] = val;
  __syncthreads();
  for (int s = 128; s > 0; s >>= 1) {
    if ((int)threadIdx.x < s) red[threadIdx.x] += red[threadIdx.x + s];
    __syncthreads();
  }
  if (threadIdx.x == 0) atomicAdd(target, red[0] * scale);
}

__global__ void init_diff(float* diff) {
  if (threadIdx.x == 0) *diff = 0.f;
}

// ---------------------------------------------------------------------------
// fq21: h[B,512] viewed as [B,64] vectors of dim 8 (stride 64 within a row).
// In-place quantize against embed[8,NEC]; accumulate mean-sq error.
// ---------------------------------------------------------------------------
__global__ __launch_bounds__(256)
void fq21_quant(float* __restrict__ hq, const float* __restrict__ embed,
                float* __restrict__ diff) {
  __shared__ float sE[8 * NEC];   // 16 KB codebook, broadcast reads
  __shared__ float sN[NEC];
  for (int i = threadIdx.x; i < 8 * NEC; i += 256) sE[i] = embed[i];
  __syncthreads();
  for (int j = threadIdx.x; j < NEC; j += 256) {
    float s = 0.f;
#pragma unroll
    for (int d = 0; d < 8; ++d) { float e = sE[d * NEC + j]; s = fmaf(e, e, s); }
    sN[j] = s;
  }
  __syncthreads();

  const int idx = blockIdx.x * 256 + threadIdx.x;   // [0, B*64)
  const int b = idx >> 6, c = idx & 63;
  float* base = hq + (size_t)b * 512 + c;
  float v[8];
#pragma unroll
  for (int d = 0; d < 8; ++d) v[d] = base[d * 64];

  float best = 3.4e38f; int bj = 0;
  for (int j = 0; j < NEC; ++j) {
    float dot = 0.f;
#pragma unroll
    for (int d = 0; d < 8; ++d) dot = fmaf(v[d], sE[d * NEC + j], dot);
    float score = sN[j] - 2.f * dot;                // |e|^2 - 2 v.e  (argmin)
    if (score < best) { best = score; bj = j; }
  }

  float err = 0.f;
#pragma unroll
  for (int d = 0; d < 8; ++d) {
    float e = sE[d * NEC + bj];
    float dd = e - v[d];
    err = fmaf(dd, dd, err);
    base[d * 64] = e;                               // straight-through == q
  }
  block_reduce_add(err, diff, 1.f / (float)(BSZ * 64 * 8));
}

// ---------------------------------------------------------------------------
// Weight quantization of W2: vector (g2,g1) has v[2i+o] = W2[g2*2+o, g1*8+i],
// quantized against embed_q2[16,NEC]; scatter back into qW2 same layout.
// ---------------------------------------------------------------------------
__global__ __launch_bounds__(256)
void wq2_quant(const float* __restrict__ W2, const float* __restrict__ embed,
               float* __restrict__ qW2, float* __restrict__ diff) {
  __shared__ float sE[16 * NEC];  // 32 KB
  __shared__ float sN[NEC];
  for (int i = threadIdx.x; i < 16 * NEC; i += 256) sE[i] = embed[i];
  __syncthreads();
  for (int j = threadIdx.x; j < NEC; j += 256) {
    float s = 0.f;
#pragma unroll
    for (int d = 0; d < 16; ++d) { float e = sE[d * NEC + j]; s = fmaf(e, e, s); }
    sN[j] = s;
  }
  __syncthreads();

  const int idx = blockIdx.x * 256 + threadIdx.x;   // [0, 256*64)
  const int g2 = idx >> 6, g1 = idx & 63;
  float v[16];
#pragma unroll
  for (int d = 0; d < 16; ++d)
    v[d] = W2[(size_t)(g2 * 2 + (d & 1)) * 512 + g1 * 8 + (d >> 1)];

  float best = 3.4e38f; int bj = 0;
  for (int j = 0; j < NEC; ++j) {
    float dot = 0.f;
#pragma unroll
    for (int d = 0; d < 16; ++d) dot = fmaf(v[d], sE[d * NEC + j], dot);
    float score = sN[j] - 2.f * dot;
    if (score < best) { best = score; bj = j; }
  }

  float err = 0.f;
#pragma unroll
  for (int d = 0; d < 16; ++d) {
    float e = sE[d * NEC + bj];
    float dd = e - v[d];
    err = fmaf(dd, dd, err);
    qW2[(size_t)(g2 * 2 + (d & 1)) * 512 + g1 * 8 + (d >> 1)] = e;
  }
  block_reduce_add(err, diff, 1.f / (float)(256 * 64 * 16));
}

// ---------------------------------------------------------------------------
// BatchNorm training stats: one block handles 32 channels, threads (32 ch x 8
// row-stripes) -> coalesced 128B row segments. Biased variance, eps 1e-5.
// ---------------------------------------------------------------------------
__global__ __launch_bounds__(256)
void bn_stats(const float* __restrict__ qx, float* __restrict__ mu,
              float* __restrict__ rstd) {
  __shared__ float ss[8][32];
  __shared__ float sq[8][32];
  const int c  = threadIdx.x & 31;
  const int t  = threadIdx.x >> 5;
  const int ch = blockIdx.x * 32 + c;
  float s = 0.f, q = 0.f;
  for (int r0 = t; r0 < BSZ; r0 += 8) {
    float x = qx[(size_t)r0 * 512 + ch];
    s += x; q = fmaf(x, x, q);
  }
  ss[t][c] = s; sq[t][c] = q;
  __syncthreads();
  if (t == 0) {
#pragma unroll
    for (int u = 1; u < 8; ++u) { s += ss[u][c]; q += sq[u][c]; }
    float m   = s * (1.f / (float)BSZ);
    float var = q * (1.f / (float)BSZ) - m * m;
    mu[ch]   = m;
    rstd[ch] = rsqrtf(var + 1e-5f);
  }
}

__global__ __launch_bounds__(256)
void bn_apply(float* __restrict__ x, const float* __restrict__ mu,
              const float* __restrict__ rstd, const float* __restrict__ gamma,
              const float* __restrict__ beta) {
  const int i = blockIdx.x * 256 + threadIdx.x;
  const int c = i & 511;
  x[i] = (x[i] - mu[c]) * rstd[c] * gamma[c] + beta[c];
}

// ---------------------------------------------------------------------------
// fq22: h2[B,512] viewed as [B,256] vectors of dim 2 (elements c and c+256).
// ---------------------------------------------------------------------------
__global__ __launch_bounds__(256)
void fq22_quant(float* __restrict__ hq, const float* __restrict__ embed,
                float* __restrict__ diff) {
  __shared__ float sE0[NEC], sE1[NEC], sN[NEC];
  for (int j = threadIdx.x; j < NEC; j += 256) {
    float e0 = embed[j], e1 = embed[NEC + j];
    sE0[j] = e0; sE1[j] = e1;
    sN[j] = fmaf(e0, e0, e1 * e1);
  }
  __syncthreads();

  const int idx = blockIdx.x * 256 + threadIdx.x;   // [0, B*256)
  const int b = idx >> 8, c = idx & 255;
  float* p0 = hq + (size_t)b * 512 + c;
  float* p1 = p0 + 256;
  const float v0 = *p0, v1 = *p1;

  float best = 3.4e38f; int bj = 0;
  for (int j = 0; j < NEC; ++j) {
    float score = sN[j] - 2.f * fmaf(v0, sE0[j], v1 * sE1[j]);
    if (score < best) { best = score; bj = j; }
  }
  const float e0 = sE0[bj], e1 = sE1[bj];
  *p0 = e0; *p1 = e1;
  const float d0 = e0 - v0, d1 = e1 - v1;
  block_reduce_add(fmaf(d0, d0, d1 * d1), diff, 1.f / (float)(BSZ * 256 * 2));
}

// ---------------------------------------------------------------------------
extern "C" void kernel_launch(void* const* d_in, const int* in_sizes, int n_in,
                              void* d_out, int out_size, void* d_ws, size_t ws_size,
                              hipStream_t stream) {
  (void)in_sizes; (void)n_in; (void)out_size; (void)ws_size;

  const float* x     = (const float*)d_in[0];   // [4096, 784]
  const float* W1    = (const float*)d_in[1];   // [512, 784]
  const float* W2    = (const float*)d_in[2];   // [512, 512]
  const float* W3    = (const float*)d_in[3];   // [10, 512]
  const float* gamma = (const float*)d_in[4];   // [512]
  const float* beta  = (const float*)d_in[5];   // [512]
  const float* Ef21  = (const float*)d_in[6];   // [8, 512]
  const float* Eq2   = (const float*)d_in[7];   // [16, 512]
  const float* Ef22  = (const float*)d_in[8];   // [2, 512]

  float* out  = (float*)d_out;                  // [4096,10] then scalar diff
  float* diff = out + (size_t)BSZ * 10;

  float* buf0 = (float*)d_ws;                   // h -> qx -> xn   (in-place)
  float* buf1 = buf0 + (size_t)BSZ * 512;       // h2 -> qx2       (in-place)
  float* qW2  = buf1 + (size_t)BSZ * 512;       // [512*512]
  float* mu   = qW2 + 512 * 512;                // [512]
  float* rstd = mu + 512;                       // [512]

  init_diff<<<1, 64, 0, stream>>>(diff);

  // fc1 + relu : h = relu(x @ W1^T)            M=4096 N=512 K=784, NT=4
  gemm_wmma_f32<true, 4, 16><<<dim3(512 / 64, BSZ / 128), 256, 0, stream>>>(
      x, W1, buf0, BSZ, 512, 784);

  // quantize fc2 weights (independent of activations)
  wq2_quant<<<(256 * 64) / 256, 256, 0, stream>>>(W2, Eq2, qW2, diff);

  // fq21 (in-place on h -> qx)
  fq21_quant<<<(BSZ * 64) / 256, 256, 0, stream>>>(buf0, Ef21, diff);

  // bn1 training-mode stats + apply (in-place qx -> xn)
  bn_stats<<<512 / 32, 256, 0, stream>>>(buf0, mu, rstd);
  bn_apply<<<(BSZ * 512) / 256, 256, 0, stream>>>(buf0, mu, rstd, gamma, beta);

  // fc2 + relu : h2 = relu(xn @ qW2^T)         M=4096 N=512 K=512, NT=4
  gemm_wmma_f32<true, 4, 16><<<dim3(512 / 64, BSZ / 128), 256, 0, stream>>>(
      buf0, qW2, buf1, BSZ, 512, 512);

  // fq22 (in-place on h2 -> qx2)
  fq22_quant<<<(BSZ * 256) / 256, 256, 0, stream>>>(buf1, Ef22, diff);

  // fc3 : out = qx2 @ W3^T                     M=4096 N=10 (padded tile) K=512
  gemm_wmma_f32<false, 1, 10><<<dim3(1, BSZ / 128), 256, 0, stream>>>(
      buf1, W3, out, BSZ, 10, 512);
}